// AdaptivePooling2D_48146583388556
// MI455X (gfx1250) — compile-verified
//
#include <hip/hip_runtime.h>

typedef float v2f __attribute__((ext_vector_type(2)));
typedef float v4f __attribute__((ext_vector_type(4)));
typedef float v8f __attribute__((ext_vector_type(8)));

#define NB   16
#define HW   224
#define CH   128
#define BINS 7
#define BIN  32   // 224 / 7

__global__ __launch_bounds__(256) void
AdaptivePooling2D_48146583388556_kernel(const float* __restrict__ x,
                                        float* __restrict__ out) {
    const int bid = blockIdx.x;                 // enumerates (b, oi, oj) = 16*7*7 = 784
    const int b   = bid / (BINS * BINS);
    const int r   = bid % (BINS * BINS);
    const int oi  = r / BINS;
    const int oj  = r % BINS;
    const int hs  = oi * BIN;
    const int ws  = oj * BIN;

    const int t    = threadIdx.x;
    const int cg   = t & 31;   // channel group: channels 4*cg .. 4*cg+3
    const int so   = t >> 5;   // wave id 0..7; also column-offset group

    // Base of this tile: row (b*224 + hs), column ws, channel 0
    const size_t rowStride = (size_t)HW * CH;   // floats per image row
    const float* base = x + ((size_t)b * HW + hs) * rowStride + (size_t)ws * CH;

    v4f acc0 = (v4f)0.0f, acc1 = (v4f)0.0f, acc2 = (v4f)0.0f, acc3 = (v4f)0.0f;

    // Each thread sums rows m=0..31 at columns ws+{so, so+8, so+16, so+24},
    // channels 4cg..4cg+3. Wave (32 lanes, fixed so) reads 512 contiguous
    // bytes per load slot -> fully coalesced global_load_b128, NT-hinted.
    #pragma unroll 4
    for (int m = 0; m < BIN; ++m) {
        const v4f* r4 = (const v4f*)(base + (size_t)m * rowStride);
        const int o = so * 32 + cg;             // (col offset)*CH/4 + cg
        acc0 += __builtin_nontemporal_load(&r4[o]);
        acc1 += __builtin_nontemporal_load(&r4[o + 256]);   //  +8 cols
        acc2 += __builtin_nontemporal_load(&r4[o + 512]);   // +16 cols
        acc3 += __builtin_nontemporal_load(&r4[o + 768]);   // +24 cols
    }
    v4f acc = (acc0 + acc1) + (acc2 + acc3);

    // Stage per-wave partials: part[so][channel]
    __shared__ float part[8][CH];
    *(v4f*)&part[so][cg * 4] = acc;
    __syncthreads();

    // Final 8-way reduction via V_WMMA_F32_16X16X4_F32 with A = ones:
    //   D[m,n] = sum_k B[k,n] (+ C[m,n])  -> column sums, replicated per row.
    // Wave w handles channels [16w, 16w+16). Column n = lane & 15 (B shares
    // the C/D column->lane mapping); lanes 0-15 supply two K rows, lanes
    // 16-31 the other two; with all-ones A the K permutation is irrelevant.
    const int lane = cg;            // t & 31
    const int w    = so;            // wave id
    const int half = lane >> 4;
    const int n    = lane & 15;
    const int chn  = 16 * w + n;

    v2f a;  a.x = 1.0f; a.y = 1.0f;

    v2f b0, b1;
    b0.x = part[2 * half + 0][chn];
    b0.y = part[2 * half + 1][chn];
    b1.x = part[2 * half + 4][chn];
    b1.y = part[2 * half + 5][chn];

    v8f c = (v8f)0.0f;
    v8f d = __builtin_amdgcn_wmma_f32_16x16x4_f32(
        /*neg_a=*/false, a, /*neg_b=*/false, b0,
        /*c_mod=*/(short)0, c, /*reuse_a=*/false, /*reuse_b=*/false);
    d = __builtin_amdgcn_wmma_f32_16x16x4_f32(
        false, a, false, b1, (short)0, d, false, false);

    // D row 0 lives in result element 0 on lanes 0-15 (N = lane).
    if (lane < 16) {
        out[(size_t)bid * CH + chn] = d[0] * (1.0f / 1024.0f);
    }
}

extern "C" void kernel_launch(void* const* d_in, const int* in_sizes, int n_in,
                              void* d_out, int out_size, void* d_ws, size_t ws_size,
                              hipStream_t stream) {
    (void)in_sizes; (void)n_in; (void)out_size; (void)d_ws; (void)ws_size;
    const float* x = (const float*)d_in[0];
    float* out = (float*)d_out;

    dim3 grid(NB * BINS * BINS);   // 784 blocks: one per (b, oi, oj) tile
    dim3 block(256);               // 8 waves (wave32)
    AdaptivePooling2D_48146583388556_kernel<<<grid, block, 0, stream>>>(x, out);
}